// STVMCache_29429115912893
// MI455X (gfx1250) — compile-verified
//
#include <hip/hip_runtime.h>
#include <math.h>

#define STVM_SIZE     65536
#define PATTERN_DIM   2048
#define MAX_PATTERNS  100

typedef __attribute__((ext_vector_type(2))) float v2f;
typedef __attribute__((ext_vector_type(4))) float v4f;
typedef __attribute__((ext_vector_type(8))) float v8f;

// ---------------------------------------------------------------------------
// Kernel 0: ||q|| (single block reduction over 2048 elements)
// ---------------------------------------------------------------------------
__global__ void stvm_qnorm_kernel(const float* __restrict__ q,
                                  float* __restrict__ qn) {
    __shared__ float red[256];
    const int tid = threadIdx.x;
    float acc = 0.0f;
    for (int i = tid; i < PATTERN_DIM; i += 256) {
        float v = q[i];
        acc = fmaf(v, v, acc);
    }
    red[tid] = acc;
    __syncthreads();
    for (int s = 128; s > 0; s >>= 1) {
        if (tid < s) red[tid] += red[tid + s];
        __syncthreads();
    }
    if (tid == 0) *qn = sqrtf(red[0]);
}

// ---------------------------------------------------------------------------
// Kernel 1: sims[i] = (p_i . q) / max(||p_i|| * ||q||, eps) / TEMPERATURE
// Block: 256 threads (8 waves), owns 128 rows. Each wave: 16 rows via
// V_WMMA_F32_16X16X4_F32, accumulating over K = 2048 (512 WMMAs).
// Pattern tiles staged memory->LDS with GLOBAL_LOAD_ASYNC_TO_LDS_B128
// (ASYNCcnt-tracked, no VGPR round trip, all 8 transfers in flight),
// drained by a single s_wait_asynccnt before the workgroup barrier.
// Pad stride 68 -> conflict-free b64 A-fragment reads, 16B-aligned stores.
// ---------------------------------------------------------------------------
__global__ void stvm_sims_kernel(const float* __restrict__ patterns,
                                 const float* __restrict__ query,
                                 const unsigned char* __restrict__ empty_slots,
                                 const float* __restrict__ qn_ptr,
                                 float* __restrict__ sims) {
    __shared__ float qlds[PATTERN_DIM];      // 8 KB: full query
    __shared__ float tile[128 * 68];         // 34 KB: 128 rows x 64 K (pad 68)
    __shared__ float wdots[8][16];           // per-wave dot results

    const int tid  = threadIdx.x;
    const int lane = tid & 31;
    const int wave = tid >> 5;
    const int half = lane >> 4;              // 0: K={0,1}, 1: K={2,3}
    const int mrow = lane & 15;              // A-matrix row M = lane % 16
    const int rowBase = blockIdx.x * 128;

    // Preload the full query into LDS (coalesced float4).
    for (int i = tid * 4; i < PATTERN_DIM; i += 256 * 4) {
        *(v4f*)&qlds[i] = *(const v4f*)&query[i];
    }

    v8f  c    = {};      // 16x16 f32 accumulator (dots; all N columns equal)
    float nacc = 0.0f;   // per-lane partial sum of squares (one K-half)

    const float* tA = &tile[(wave * 16 + mrow) * 68];
    const int lrow = tid >> 4;               // 0..15
    const int lcol = (tid & 15) * 4;         // 0..60

    for (int t = 0; t < PATTERN_DIM / 64; ++t) {
        const int k0 = t * 64;
        __syncthreads();   // protect tile reuse (and qlds on first pass)

        // Async tile fill: 128 rows x 64 floats, b128 per thread x 8,
        // all in flight at once (no VGPR destination, ASYNCcnt-tracked).
        #pragma unroll
        for (int r = 0; r < 8; ++r) {
            const int row = r * 16 + lrow;
            const float* gsrc =
                &patterns[(size_t)(rowBase + row) * PATTERN_DIM + k0 + lcol];
            // Generic shared pointer low 32 bits == LDS byte offset.
            unsigned lds_dst = (unsigned)(size_t)&tile[row * 68 + lcol];
            asm volatile("global_load_async_to_lds_b128 %0, %1, off"
                         :: "v"(lds_dst), "v"(gsrc)
                         : "memory");
        }
        asm volatile("s_wait_asynccnt 0x0" ::: "memory");
        __syncthreads();

        #pragma unroll
        for (int kk = 0; kk < 64; kk += 4) {
            const int ks = kk + half * 2;
            v2f a, b;
            a.x = tA[ks];                 // A[M=mrow, K=ks]   (VGPR0)
            a.y = tA[ks + 1];             // A[M=mrow, K=ks+1] (VGPR1)
            b.x = qlds[k0 + ks];          // B[K=ks,  N=*] = q[k]
            b.y = qlds[k0 + ks + 1];      // B[K=ks+1,N=*] = q[k+1]
            // D = A x B + C  (f32, 16x16x4)
            c = __builtin_amdgcn_wmma_f32_16x16x4_f32(
                    false, a, false, b, (short)0, c, false, false);
            nacc = fmaf(a.x, a.x, nacc);
            nacc = fmaf(a.y, a.y, nacc);
        }
    }

    // Combine the two K-half norm partials (lane L <-> lane L^16).
    nacc += __shfl_xor(nacc, 16, 32);

    // D layout (32-bit 16x16): VGPR v, lanes 0-15 -> M=v, lanes 16-31 -> M=v+8.
    // Take N=0 column: lane 0 has rows 0..7, lane 16 has rows 8..15.
    if ((lane & 15) == 0) {
        #pragma unroll
        for (int v = 0; v < 8; ++v) wdots[wave][half * 8 + v] = c[v];
    }
    __syncthreads();

    if (lane < 16) {
        const int row = rowBase + wave * 16 + lane;
        const float qn = *qn_ptr;
        const float pn = sqrtf(nacc);
        float s = wdots[wave][lane] / fmaxf(pn * qn, 1e-8f) * 10.0f; // /0.1
        if (empty_slots[row]) s = -INFINITY;
        sims[row] = s;
    }
}

// ---------------------------------------------------------------------------
// Kernel 2: iterative top-100 argmax over sims (single 1024-thread block).
// Writes out_sims (thresholded at 0), plus idx/hit to workspace.
// sims is consumed destructively (rewritten fully by kernel 1 each launch).
// ---------------------------------------------------------------------------
__global__ void stvm_topk_kernel(float* __restrict__ sims,
                                 float* __restrict__ out_sims,
                                 int* __restrict__ idx_ws,
                                 int* __restrict__ hit_ws) {
    __shared__ float sVal[1024];
    __shared__ int   sIdx[1024];
    const int tid = threadIdx.x;

    for (int k = 0; k < MAX_PATTERNS; ++k) {
        float best = -INFINITY;
        int   bi   = STVM_SIZE;
        for (int i = tid; i < STVM_SIZE; i += 1024) {
            float v = sims[i];
            if (v > best || (v == best && i < bi)) { best = v; bi = i; }
        }
        sVal[tid] = best;
        sIdx[tid] = bi;
        __syncthreads();
        for (int s = 512; s > 0; s >>= 1) {
            if (tid < s) {
                float v2 = sVal[tid + s];
                int   i2 = sIdx[tid + s];
                if (v2 > sVal[tid] || (v2 == sVal[tid] && i2 < sIdx[tid])) {
                    sVal[tid] = v2;
                    sIdx[tid] = i2;
                }
            }
            __syncthreads();
        }
        if (tid == 0) {
            const float bv = sVal[0];
            const int bidx = sIdx[0];
            const int hit  = (bv >= 0.0f) ? 1 : 0;
            out_sims[k] = hit ? bv : 0.0f;
            idx_ws[k]   = (bidx < STVM_SIZE) ? bidx : 0;
            hit_ws[k]   = hit;
            if (bidx < STVM_SIZE) sims[bidx] = -INFINITY;
        }
        __syncthreads();
    }
}

// ---------------------------------------------------------------------------
// Kernel 3: gather selected pattern rows (or zeros) into out_patterns.
// ---------------------------------------------------------------------------
__global__ void stvm_gather_kernel(const float* __restrict__ patterns,
                                   const int* __restrict__ idx_ws,
                                   const int* __restrict__ hit_ws,
                                   float* __restrict__ out_patterns) {
    const int k   = blockIdx.x;
    const int idx = idx_ws[k];
    const int hit = hit_ws[k];
    const float* src = &patterns[(size_t)idx * PATTERN_DIM];
    float* dst = &out_patterns[(size_t)k * PATTERN_DIM];
    for (int j = threadIdx.x * 4; j < PATTERN_DIM; j += 256 * 4) {
        v4f v = {};
        if (hit) v = *(const v4f*)&src[j];
        *(v4f*)&dst[j] = v;
    }
}

// ---------------------------------------------------------------------------
extern "C" void kernel_launch(void* const* d_in, const int* in_sizes, int n_in,
                              void* d_out, int out_size, void* d_ws, size_t ws_size,
                              hipStream_t stream) {
    const float*         query    = (const float*)d_in[0];          // [2048]
    const float*         patterns = (const float*)d_in[1];          // [65536, 2048]
    const unsigned char* empty    = (const unsigned char*)d_in[2];  // [65536] bool

    float* out_patterns = (float*)d_out;                            // [100, 2048]
    float* out_sims     = out_patterns + MAX_PATTERNS * PATTERN_DIM; // [100]

    // Workspace layout
    float* sims = (float*)d_ws;                 // 65536 floats
    float* qn   = sims + STVM_SIZE;             // 1 float
    int*   idxs = (int*)(qn + 1);               // 100 ints
    int*   hits = idxs + MAX_PATTERNS;          // 100 ints

    stvm_qnorm_kernel<<<1, 256, 0, stream>>>(query, qn);
    stvm_sims_kernel<<<STVM_SIZE / 128, 256, 0, stream>>>(patterns, query, empty, qn, sims);
    stvm_topk_kernel<<<1, 1024, 0, stream>>>(sims, out_sims, idxs, hits);
    stvm_gather_kernel<<<MAX_PATTERNS, 256, 0, stream>>>(patterns, idxs, hits, out_patterns);
}